// NODE_TEXTCNN_PARALLEL_2980707303909
// MI455X (gfx1250) — compile-verified
//
#include <hip/hip_runtime.h>
#include <hip/hip_bf16.h>

// ---------------------------------------------------------------------------
// Shapes
// ---------------------------------------------------------------------------
#define BB   64
#define SS   512
#define DD   256
#define NCC  10

typedef __attribute__((ext_vector_type(2))) float v2f;
typedef __attribute__((ext_vector_type(8))) float v8f;

// RK4 single-step series coefficients for h = 0.1
#define CO_H   0.1f
#define CO_H2  0.005f            // h^2/2
#define CO_H3  1.6666667e-4f     // h^3/6
#define CO_H4  4.1666668e-6f     // h^4/24
#define INV_STD 0.999995f        // 1/sqrt(1 + 1e-5)

// ---------------------------------------------------------------------------
// Setup kernels (tiny)
// ---------------------------------------------------------------------------

// X[(b*S+s)*D + d] = emb[ids[b*S+s]*D + d], vectorized float4
__global__ void gather_x(float4* __restrict__ X4, const float4* __restrict__ emb4,
                         const int* __restrict__ ids, int total4) {
  int i = blockIdx.x * 256 + threadIdx.x;
  if (i < total4) {
    int r = i >> 6;           // 64 float4 per row of 256 floats
    int c = i & 63;
    X4[i] = emb4[(size_t)ids[r] * 64 + c];
  }
}

// zero the 4-row overread pad after X (1024 floats)
__global__ void pad_zero(float4* __restrict__ p) {
  p[threadIdx.x] = make_float4(0.f, 0.f, 0.f, 0.f);
}

// w (O=128, D=256, KW) -> Wr (O, KW*256) with layout [o][j*256 + d]
__global__ void reorder_w(float* __restrict__ dst, const float* __restrict__ src, int KW) {
  int i = blockIdx.x * 256 + threadIdx.x;
  int tot = 128 * 256 * KW;
  if (i < tot) {
    int o   = i / (256 * KW);
    int rem = i - o * 256 * KW;   // = d*KW + j
    int d   = rem / KW;
    int j   = rem - d * KW;
    dst[(size_t)o * (256 * KW) + j * 256 + d] = src[i];
  }
}

// C = A * B, all 256x256 row-major
__global__ void matmul256(float* __restrict__ C, const float* __restrict__ A,
                          const float* __restrict__ B) {
  int col = blockIdx.x * 16 + threadIdx.x;
  int row = blockIdx.y * 16 + threadIdx.y;
  float s = 0.0f;
  #pragma unroll 8
  for (int k = 0; k < 256; ++k)
    s = fmaf(A[row * 256 + k], B[k * 256 + col], s);
  C[row * 256 + col] = s;
}

// Phi = I + h*W + h2*W^2 + h3*W^3 + h4*W^4 (elementwise from precomputed powers)
__global__ void phi_build(float* __restrict__ Phi, const float* __restrict__ W,
                          const float* __restrict__ P2, const float* __restrict__ P3,
                          const float* __restrict__ P4) {
  int i = blockIdx.x * 256 + threadIdx.x;   // 65536 total
  float v = CO_H * W[i] + CO_H2 * P2[i] + CO_H3 * P3[i] + CO_H4 * P4[i];
  if ((i >> 8) == (i & 255)) v += 1.0f;
  Phi[i] = v;
}

// d = h*b + h2*W b + h3*W^2 b + h4*W^3 b ; also seed cA = d
__global__ void dvec_build(float* __restrict__ dv, float* __restrict__ cA,
                           const float* __restrict__ bvec, const float* __restrict__ W,
                           const float* __restrict__ P2, const float* __restrict__ P3) {
  int n = threadIdx.x;   // 256 threads
  float s1 = 0.f, s2 = 0.f, s3 = 0.f;
  for (int k = 0; k < 256; ++k) {
    float bk = bvec[k];
    s1 = fmaf(W [n * 256 + k], bk, s1);
    s2 = fmaf(P2[n * 256 + k], bk, s2);
    s3 = fmaf(P3[n * 256 + k], bk, s3);
  }
  float v = CO_H * bvec[n] + CO_H2 * s1 + CO_H3 * s2 + CO_H4 * s3;
  dv[n] = v;
  cA[n] = v;
}

// dst = Phi * src + dv   (256x256 matvec, one block of 256 threads)
__global__ void matvec_add(float* __restrict__ dst, const float* __restrict__ Phi,
                           const float* __restrict__ src, const float* __restrict__ dv) {
  int n = threadIdx.x;
  float s = dv[n];
  for (int k = 0; k < 256; ++k)
    s = fmaf(Phi[n * 256 + k], src[k], s);
  dst[n] = s;
}

// ---------------------------------------------------------------------------
// WMMA GEMM + fused max-pool.
//   For batch b and 16-wide output panel o0:
//     C[t, n] = sum_k patch(b, t, k) * Wb[o0+n][k]      (K = KW*256)
//   then epilogue (BN+ReLU for conv / +c for ODE) and max over t.
//
//   Each 4-wave block owns all 32 seq tiles; each wave owns 8 tiles that it
//   advances through K *together*: one B fragment load feeds 8 independent
//   WMMA accumulator chains (8-way ILP, 9 loads per 8 WMMAs). The 8 A-tile
//   addresses differ by a constant 0x10000 bytes -> immediate offsets.
//   Rows beyond S read into a zeroed 4-row pad; their outputs are masked.
//   Max pool reduces in registers -> shuffle -> LDS. No atomics.
//
// A-matrix f32 16x4: lane<16 -> M=lane, K={0,1}; lane>=16 -> M=lane-16, K={2,3}
// B-matrix f32 4x16: lane<16 -> N=lane, K={0,1}; lane>=16 -> N=lane-16, K={2,3}
// C 16x16 f32:       VGPR e -> M = e + (lane>=16 ? 8 : 0), N = lane&15
// ---------------------------------------------------------------------------
template <int KW, bool IS_ODE, int OTOT>
__global__ __launch_bounds__(128) void gemm_pool(
    const float* __restrict__ X,      // (B*S + 4, 256), 4 pad rows
    const float* __restrict__ Wb,     // (OTOT, KW*256) row-major
    const float* __restrict__ bias,   // conv: (OTOT)
    const float* __restrict__ gamma,  // conv: (OTOT)
    const float* __restrict__ beta,   // conv: (OTOT)
    const float* __restrict__ cvec,   // ode : (256) composed bias
    float* __restrict__ out,          // (B, OTOT) pooled
    int Tvalid) {
  constexpr int KD = KW * 256;
  const int b    = blockIdx.x;
  const int o0   = blockIdx.y * 16;
  const int tid  = threadIdx.x;
  const int wave = tid >> 5;
  const int lane = tid & 31;
  const int m     = lane & 15;            // A row within tile / B column
  const int kbase = (lane >> 4) * 2;      // K sub-offset {0 or 2}
  const int mofs  = (lane >> 4) * 8;      // C M offset for this lane group

  // A base for this lane: row (wave*16 + m) of batch b; chunk offset is
  // exactly kg = j*256 + d, and tile i adds i*64 rows = i*16384 floats.
  const float* Abase = X + ((size_t)b * SS + wave * 16 + m) * DD;
  const float* Brow  = Wb + (size_t)(o0 + m) * KD;

  float gg = 1.f, bb = 0.f, be = 0.f;
  if constexpr (!IS_ODE) {
    bb = bias[o0 + m];
    gg = gamma[o0 + m] * INV_STD;
    be = beta[o0 + m];
  }

  v8f acc[8];
  #pragma unroll
  for (int i = 0; i < 8; ++i) acc[i] = (v8f){};

  #pragma unroll 1
  for (int kc = 0; kc < KD / 4; ++kc) {
    const int kg = kc * 4 + kbase;        // even -> float2 loads stay aligned
    const float2 b2 = *(const float2*)(Brow + kg);
    v2f bv; bv[0] = b2.x; bv[1] = b2.y;
    #pragma unroll
    for (int i = 0; i < 8; ++i) {
      const float2 a2 = *(const float2*)(Abase + (size_t)i * 16384 + kg);
      v2f av; av[0] = a2.x; av[1] = a2.y;
      acc[i] = __builtin_amdgcn_wmma_f32_16x16x4_f32(
          false, av, false, bv, (short)0, acc[i], false, false);
    }
  }

  float maxv = IS_ODE ? -3.4e38f : 0.0f;
  #pragma unroll
  for (int i = 0; i < 8; ++i) {
    const int t0 = wave * 16 + i * 64;    // tile (wave + 4*i)
    #pragma unroll
    for (int e = 0; e < 8; ++e) {
      const float v = acc[i][e];
      if constexpr (IS_ODE) {
        maxv = fmaxf(maxv, v);
      } else {
        const int t = t0 + e + mofs;
        if (t < Tvalid) {
          float val = fmaf(gg, v + bb, be);   // gamma*((c+b)*inv_std)+beta
          val = fmaxf(val, 0.0f);             // ReLU
          maxv = fmaxf(maxv, val);            // max over positions
        }
      }
    }
  }

  // lanes L and L+16 hold the same output column n = lane&15 -> pair reduce
  maxv = fmaxf(maxv, __shfl_xor(maxv, 16, 32));

  __shared__ float red[4 * 16];
  if (lane < 16) red[wave * 16 + lane] = maxv;
  __syncthreads();
  if (tid < 16) {
    float r = fmaxf(fmaxf(red[tid], red[16 + tid]),
                    fmaxf(red[32 + tid], red[48 + tid]));
    if constexpr (IS_ODE) r += cvec[o0 + tid];   // max(z*M + c) = max(z*M) + c
    out[(size_t)b * OTOT + o0 + tid] = r;
  }
}

// ---------------------------------------------------------------------------
// Final classifier: out[b][c] = bc[c] + feats(b,:) . wc[c,:]
// feats = concat(node(256), p3(128), p4(128), p5(128))
// ---------------------------------------------------------------------------
__global__ void classifier(float* __restrict__ out, const float* __restrict__ node,
                           const float* __restrict__ p3, const float* __restrict__ p4,
                           const float* __restrict__ p5, const float* __restrict__ wc,
                           const float* __restrict__ bc) {
  int tid = threadIdx.x;
  if (tid >= BB * NCC) return;
  int b = tid / NCC, c = tid % NCC;
  const float* row = wc + (size_t)c * 640;
  float s = bc[c];
  for (int k = 0; k < 256; ++k) s = fmaf(node[b * 256 + k], row[k], s);
  for (int k = 0; k < 128; ++k) s = fmaf(p3[b * 128 + k], row[256 + k], s);
  for (int k = 0; k < 128; ++k) s = fmaf(p4[b * 128 + k], row[384 + k], s);
  for (int k = 0; k < 128; ++k) s = fmaf(p5[b * 128 + k], row[512 + k], s);
  out[b * NCC + c] = s;
}

// ---------------------------------------------------------------------------
// Launch
// ---------------------------------------------------------------------------
extern "C" void kernel_launch(void* const* d_in, const int* in_sizes, int n_in,
                              void* d_out, int out_size, void* d_ws, size_t ws_size,
                              hipStream_t stream) {
  const int*   ids = (const int*)  d_in[0];
  const float* emb = (const float*)d_in[1];
  const float* W   = (const float*)d_in[2];
  const float* bo  = (const float*)d_in[3];
  const float* w3  = (const float*)d_in[4];
  const float* b3  = (const float*)d_in[5];
  const float* g3  = (const float*)d_in[6];
  const float* be3 = (const float*)d_in[7];
  const float* w4  = (const float*)d_in[8];
  const float* b4  = (const float*)d_in[9];
  const float* g4  = (const float*)d_in[10];
  const float* be4 = (const float*)d_in[11];
  const float* w5  = (const float*)d_in[12];
  const float* b5  = (const float*)d_in[13];
  const float* g5  = (const float*)d_in[14];
  const float* be5 = (const float*)d_in[15];
  const float* wc  = (const float*)d_in[16];
  const float* bc  = (const float*)d_in[17];

  float* ws = (float*)d_ws;
  const size_t XN = (size_t)BB * SS * DD;     // 8,388,608
  float* X    = ws;                  size_t off = XN;
  float* Xpad = ws + off; off += 1024;        // 4 overread rows
  float* P2   = ws + off; off += 65536;
  float* P3   = ws + off; off += 65536;
  float* P4   = ws + off; off += 65536;
  float* Phi  = ws + off; off += 65536;
  float* G2   = ws + off; off += 65536;
  float* G4   = ws + off; off += 65536;
  float* G8   = ws + off; off += 65536;
  float* G10  = ws + off; off += 65536;
  float* dv   = ws + off; off += 256;
  float* cA   = ws + off; off += 256;
  float* cB   = ws + off; off += 256;
  float* Wr3  = ws + off; off += 128 * 768;
  float* Wr4  = ws + off; off += 128 * 1024;
  float* Wr5  = ws + off; off += 128 * 1280;
  float* node = ws + off; off += BB * 256;
  float* p3b  = ws + off; off += BB * 128;
  float* p4b  = ws + off; off += BB * 128;
  float* p5b  = ws + off; off += BB * 128;

  // 1) gather embeddings -> X, zero the pad rows
  gather_x<<<(int)(XN / 4 / 256), 256, 0, stream>>>(
      (float4*)X, (const float4*)emb, ids, (int)(XN / 4));
  pad_zero<<<1, 256, 0, stream>>>((float4*)Xpad);

  // 2) reorder conv weights to (o, j*256+d)
  reorder_w<<<(128 * 256 * 3 + 255) / 256, 256, 0, stream>>>(Wr3, w3, 3);
  reorder_w<<<(128 * 256 * 4 + 255) / 256, 256, 0, stream>>>(Wr4, w4, 4);
  reorder_w<<<(128 * 256 * 5 + 255) / 256, 256, 0, stream>>>(Wr5, w5, 5);

  // 3) compose the affine RK4 map: Phi = I + hW + h2 W^2 + h3 W^3 + h4 W^4
  dim3 g16(16, 16), b16(16, 16);
  matmul256<<<g16, b16, 0, stream>>>(P2, W, W);
  matmul256<<<g16, b16, 0, stream>>>(P3, P2, W);
  matmul256<<<g16, b16, 0, stream>>>(P4, P3, W);
  phi_build<<<256, 256, 0, stream>>>(Phi, W, P2, P3, P4);
  // Phi^10 by squaring: G10 = ((Phi^2)^2)^2 * Phi^2
  matmul256<<<g16, b16, 0, stream>>>(G2, Phi, Phi);
  matmul256<<<g16, b16, 0, stream>>>(G4, G2, G2);
  matmul256<<<g16, b16, 0, stream>>>(G8, G4, G4);
  matmul256<<<g16, b16, 0, stream>>>(G10, G8, G2);
  // composed bias c = sum_{j=0..9} Phi^j d
  dvec_build<<<1, 256, 0, stream>>>(dv, cA, bo, W, P2, P3);
  float* csrc = cA; float* cdst = cB;
  for (int i = 0; i < 9; ++i) {
    matvec_add<<<1, 256, 0, stream>>>(cdst, Phi, csrc, dv);
    float* t = csrc; csrc = cdst; cdst = t;
  }
  // csrc now holds the final composed bias

  // 4) ODE GEMM + seq-max pool:  node[b,n] = max_s (X Phi10^T)[s,n] + c[n]
  gemm_pool<1, true, 256><<<dim3(BB, 16), 128, 0, stream>>>(
      X, G10, nullptr, nullptr, nullptr, csrc, node, SS);

  // 5) TextCNN branches (implicit GEMM + BN + ReLU + max pool)
  gemm_pool<3, false, 128><<<dim3(BB, 8), 128, 0, stream>>>(
      X, Wr3, b3, g3, be3, nullptr, p3b, SS - 2);
  gemm_pool<4, false, 128><<<dim3(BB, 8), 128, 0, stream>>>(
      X, Wr4, b4, g4, be4, nullptr, p4b, SS - 3);
  gemm_pool<5, false, 128><<<dim3(BB, 8), 128, 0, stream>>>(
      X, Wr5, b5, g5, be5, nullptr, p5b, SS - 4);

  // 6) classifier
  classifier<<<1, BB * NCC, 0, stream>>>((float*)d_out, node, p3b, p4b, p5b, wc, bc);
}